// BatchedCholeskyOrthogonalization_19928648254159
// MI455X (gfx1250) — compile-verified
//
#include <hip/hip_runtime.h>

// CDNA5 / gfx1250: fp32 WMMA 16x16x4. Wave32.
typedef __attribute__((ext_vector_type(2))) float v2f;
typedef __attribute__((ext_vector_type(4))) float v4f;
typedef __attribute__((ext_vector_type(8))) float v8f;

#define NN 128           // rows of X, size of S
#define MM 256           // cols of X
#define XS_STRIDE 260    // LDS row stride for X/W (floats); 260*4 B is 16B aligned, bank step 4
#define SS_STRIDE 132    // LDS row stride for S/L (floats); bank step 4
#define EPS_SCALE 0.001f
#define THREADS 512      // 16 waves; one 200KB-LDS workgroup per WGP anyway
#define NWAVES  16

__global__ __launch_bounds__(THREADS, 1)
void chol_ortho_gfx1250(const float* __restrict__ X, float* __restrict__ Wout)
{
    extern __shared__ float lds[];
    float* Xs  = lds;                        // 128 x 260  (X, later W, in place)
    float* Ss  = lds + NN * XS_STRIDE;       // 128 x 132  (S lower tri, later L, in place)
    float* red = Ss + NN * SS_STRIDE;        // 1 reduction cell

    const int tid  = threadIdx.x;
    const int lane = tid & 31;
    const int wave = tid >> 5;
    const int lm   = lane & 15;   // row/col within tile
    const int lh   = lane >> 4;   // half-wave select (K split for A/B, M split for C/D)

    const int b = blockIdx.x;
    const float* Xg = X    + (size_t)b * (NN * MM);
    float*       Wg = Wout + (size_t)b * (NN * MM);

    // ---------------- Phase 0: stage X into LDS (float4, coalesced) ----------------
    for (int idx = tid; idx < NN * MM / 4; idx += THREADS) {
        const int row  = idx >> 6;      // 64 float4 per row
        const int col4 = idx & 63;
        v4f v = *(const v4f*)(Xg + row * MM + col4 * 4);
        *(v4f*)(Xs + row * XS_STRIDE + col4 * 4) = v;
    }
    if (tid == 0) red[0] = 0.0f;
    __syncthreads();

    // ---------------- Phase 1: lower triangle of S = X * X^T via v_wmma_f32_16x16x4_f32 ----
    // Only tiles (I,J) with J <= I are ever read (Cholesky + trisolve use lower + diag).
    // 36 tiles balanced over 16 waves via flat triangular index.
    // A(m,k): lane = (k%4/2)*16 + m, vgpr = k%2. B(k,n): identical gather with n (S = X X^T).
    for (int t = wave; t < 36; t += NWAVES) {
        int I = 0;
        while (((I + 1) * (I + 2)) >> 1 <= t) ++I;      // wave-uniform, SALU
        const int J = t - ((I * (I + 1)) >> 1);
        const float* arow = Xs + (I * 16 + lm) * XS_STRIDE;
        const float* brow = Xs + (J * 16 + lm) * XS_STRIDE;
        v8f c = {};
        for (int k0 = 0; k0 < MM; k0 += 4) {
            v2f a  = *(const v2f*)(arow + k0 + lh * 2);   // 8B-aligned ds_load_b64
            v2f bb = *(const v2f*)(brow + k0 + lh * 2);
            c = __builtin_amdgcn_wmma_f32_16x16x4_f32(false, a, false, bb,
                                                      (short)0, c, false, false);
        }
        // C/D layout: vgpr r, lane l -> (m = r + 8*(l/16), n = l%16)
        float* srow = Ss + (I * 16 + 8 * lh) * SS_STRIDE + J * 16 + lm;
#pragma unroll
        for (int r = 0; r < 8; ++r) srow[r * SS_STRIDE] = c[r];
    }
    __syncthreads();

    // ---------------- Phase 2: eps = mean(diag(S)) * 1e-3, add to diagonal ----------------
    if (tid < NN) atomicAdd(&red[0], Ss[tid * SS_STRIDE + tid]);   // ds_add_f32
    __syncthreads();
    const float eps = red[0] * (EPS_SCALE / (float)NN);
    if (tid < NN) Ss[tid * SS_STRIDE + tid] += eps;
    __syncthreads();

    // ---------------- Phase 3: in-place Cholesky (lower), right-looking ----------------
    for (int j = 0; j < NN; ++j) {
        const float djj = Ss[j * SS_STRIDE + j];
        const float Ljj = sqrtf(djj);
        const float inv = 1.0f / Ljj;
        __syncthreads();                       // everyone has read djj
        if (tid == 0) Ss[j * SS_STRIDE + j] = Ljj;
        for (int i = j + 1 + tid; i < NN; i += THREADS)
            Ss[i * SS_STRIDE + j] *= inv;
        __syncthreads();                       // column j scaled
        for (int i = j + 1 + wave; i < NN; i += NWAVES) {
            const float Lij = Ss[i * SS_STRIDE + j];
            for (int k = j + 1 + lane; k <= i; k += 32)
                Ss[i * SS_STRIDE + k] -= Lij * Ss[k * SS_STRIDE + j];
        }
        __syncthreads();                       // trailing update visible
    }

    // ---------------- Phase 4: solve L * W = X, blocked by 16 rows, WMMA updates ----------------
    for (int I = 0; I < 8; ++I) {
        if (I > 0) {
            // X_I -= sum_{J<I} L_IJ * W_J   (16 x 256, one 16-col tile per wave)
            const int c0 = wave * 16;
            v8f acc = {};
            for (int J = 0; J < I; ++J) {
                const float* Lrow = Ss + (I * 16 + lm) * SS_STRIDE + J * 16;
#pragma unroll
                for (int k0 = 0; k0 < 16; k0 += 4) {
                    v2f a = *(const v2f*)(Lrow + k0 + lh * 2);
                    const int kk = J * 16 + k0 + lh * 2;
                    v2f bb;
                    bb.x = Xs[(kk + 0) * XS_STRIDE + c0 + lm];
                    bb.y = Xs[(kk + 1) * XS_STRIDE + c0 + lm];
                    acc = __builtin_amdgcn_wmma_f32_16x16x4_f32(false, a, false, bb,
                                                                (short)0, acc, false, false);
                }
            }
            float* xrow = Xs + (I * 16 + 8 * lh) * XS_STRIDE + c0 + lm;
#pragma unroll
            for (int r = 0; r < 8; ++r) xrow[r * XS_STRIDE] -= acc[r];
        }
        __syncthreads();

        // 16x16 diagonal-block forward substitution; threads 0..255 own one RHS column each
        if (tid < MM) {
            const int c = tid;
            float x[16];
#pragma unroll
            for (int r = 0; r < 16; ++r) x[r] = Xs[(I * 16 + r) * XS_STRIDE + c];
#pragma unroll
            for (int r = 0; r < 16; ++r) {
                const float w = x[r] / Ss[(I * 16 + r) * SS_STRIDE + I * 16 + r];
                x[r] = w;
#pragma unroll
                for (int s = r + 1; s < 16; ++s)
                    x[s] -= Ss[(I * 16 + s) * SS_STRIDE + I * 16 + r] * w;
            }
#pragma unroll
            for (int r = 0; r < 16; ++r) Xs[(I * 16 + r) * XS_STRIDE + c] = x[r];
        }
        __syncthreads();
    }

    // ---------------- Phase 5: write W out (float4, coalesced) ----------------
    for (int idx = tid; idx < NN * MM / 4; idx += THREADS) {
        const int row  = idx >> 6;
        const int col4 = idx & 63;
        *(v4f*)(Wg + row * MM + col4 * 4) = *(const v4f*)(Xs + row * XS_STRIDE + col4 * 4);
    }
}

extern "C" void kernel_launch(void* const* d_in, const int* in_sizes, int n_in,
                              void* d_out, int out_size, void* d_ws, size_t ws_size,
                              hipStream_t stream) {
    (void)n_in; (void)d_ws; (void)ws_size; (void)out_size;
    const float* w = (const float*)d_in[0];
    float* out = (float*)d_out;
    const int B = in_sizes[0] / (NN * MM);          // 1024
    const size_t shmem = (size_t)(NN * XS_STRIDE + NN * SS_STRIDE + 4) * sizeof(float); // ~200 KB of 320 KB WGP LDS
    hipLaunchKernelGGL(chol_ortho_gfx1250, dim3(B), dim3(THREADS), shmem, stream, w, out);
}